// EnhancedLDEPooling_74783970558326
// MI455X (gfx1250) — compile-verified
//
#include <hip/hip_runtime.h>
#include <math.h>

typedef __attribute__((ext_vector_type(2))) float v2f;
typedef __attribute__((ext_vector_type(8))) float v8f;
typedef unsigned int v4u __attribute__((ext_vector_type(4)));
typedef int v4i __attribute__((ext_vector_type(4)));
typedef int v8i __attribute__((ext_vector_type(8)));

#define T_TOT 2048
#define DDIM  256
#define KCL   8
#define TCH   128        // t-rows per workgroup
#define NCH   16         // T_TOT / TCH
#define XS_STRIDE 260    // padded row stride (floats): float4-aligned, rotates banks
#define WS_STRIDE 16     // k padded 8 -> 16 (zeros) for WMMA M dimension

// TDM availability (device pass for gfx1250 only; host pass falls back)
#if defined(__gfx1250__) && defined(__has_builtin)
#  if __has_builtin(__builtin_amdgcn_tensor_load_to_lds)
#    define HAVE_TDM 1
#  endif
#  if __has_builtin(__builtin_amdgcn_s_wait_tensorcnt)
#    define HAVE_WAIT_TENSORCNT 1
#  endif
#endif

// ---------------- Pass 1: fused distance -> softmax -> WMMA accumulate ----------------
__global__ __launch_bounds__(128) void lde_pass1(
    const float* __restrict__ x, const float* __restrict__ centers,
    const float* __restrict__ scale, const float* __restrict__ temperature,
    float* __restrict__ swx_part, float* __restrict__ swx2_part,
    float* __restrict__ sw_part)
{
  extern __shared__ float smem[];
  float* xs  = smem;                       // [TCH][XS_STRIDE]
  float* ws  = xs + TCH * XS_STRIDE;       // [TCH][WS_STRIDE]
  float* cs  = ws + TCH * WS_STRIDE;       // [KCL][DDIM]
  float* c2s = cs + KCL * DDIM;            // [KCL]

  const int tid = threadIdx.x;
  const int b   = blockIdx.x >> 4;
  const int ch  = blockIdx.x & 15;
  const long t0 = (long)b * T_TOT + (long)ch * TCH;

#ifdef HAVE_TDM
  // ---- Tensor Data Mover: one DMA for the whole 128x256 f32 tile, with
  // hardware LDS padding: pad_interval=7 (256 dwords) + pad_amount=3 (4 dwords)
  // => LDS row stride 260 dwords == XS_STRIDE. Issued by wave 0 only (EXEC-
  // independent instruction; all waves issuing would duplicate the DMA).
  if (tid < 32) {
    unsigned long long gaddr = (unsigned long long)(const void*)(x + t0 * DDIM);
    unsigned lds_base = __builtin_amdgcn_groupstaticsize();  // dynamic LDS offset

    v4u g0;
    g0[0] = 1u;                                   // count=1, is_restore=0, gather=0
    g0[1] = lds_base;                             // lds_addr (bytes)
    g0[2] = (unsigned)(gaddr & 0xFFFFFFFFull);    // global_addr[31:0]
    g0[3] = (unsigned)((gaddr >> 32) & 0x1FFFFFFull) | (2u << 30);  // addr[56:32], type=2

    v8i g1;
    g1[0] = (int)((2u << 16) | (1u << 20) | (7u << 22) | (3u << 25)); // data_size=4B, pad
    g1[1] = (int)(256u << 16);                    // tensor_dim0[15:0]=256 (atomic_barrier=0)
    g1[2] = (int)(128u << 16);                    // tensor_dim0[31:16]=0, tensor_dim1[15:0]=128
    g1[3] = (int)(256u << 16);                    // tensor_dim1[31:16]=0, tile_dim0=256
    g1[4] = (int)128u;                            // tile_dim1=128, tile_dim2=0
    g1[5] = (int)256u;                            // tensor_dim0_stride[31:0]=256
    g1[6] = 0;                                    // stride[47:32]=0, dim1_stride[15:0]=0
    g1[7] = 0;

    v4i gz = {};                                  // groups 2/3 unused (2D tensor)
#if __clang_major__ >= 23
    v8i gz8 = {};
    __builtin_amdgcn_tensor_load_to_lds(g0, g1, gz, gz, gz8, 0);
#else
    __builtin_amdgcn_tensor_load_to_lds(g0, g1, gz, gz, 0);
#endif
#ifdef HAVE_WAIT_TENSORCNT
    __builtin_amdgcn_s_wait_tensorcnt(0);
#else
    asm volatile("s_wait_tensorcnt 0x0" ::: "memory");
#endif
  }
  // centers -> LDS (512 float4 total, 4 per thread), coalesced
  for (int j = 0; j < 4; ++j) {
    int idx = j * 128 + tid;
    float4 v = ((const float4*)centers)[idx];
    float* dst = &cs[idx * 4];
    dst[0] = v.x; dst[1] = v.y; dst[2] = v.z; dst[3] = v.w;
  }
#else
  // Fallback: manual coalesced staging
  for (int j = 0; j < 4; ++j) {
    int idx = j * 128 + tid;
    float4 v = ((const float4*)centers)[idx];
    float* dst = &cs[idx * 4];
    dst[0] = v.x; dst[1] = v.y; dst[2] = v.z; dst[3] = v.w;
  }
  const float4* xg = (const float4*)(x + t0 * DDIM);
  for (int j = 0; j < 64; ++j) {
    int idx = j * 128 + tid;           // idx = row*64 + col4
    int row = idx >> 6, c4 = idx & 63;
    float4 v = xg[idx];
    float* dst = &xs[row * XS_STRIDE + c4 * 4];
    dst[0] = v.x; dst[1] = v.y; dst[2] = v.z; dst[3] = v.w;
  }
#endif
  __syncthreads();

  if (tid < KCL) {                      // ||c_k||^2
    float s = 0.f;
    for (int d = 0; d < DDIM; ++d) { float c = cs[tid * DDIM + d]; s += c * c; }
    c2s[tid] = s;
  }
  __syncthreads();

  // ---- per-thread (one t-row): 8 dots + ||x||^2, softmax over K ----
  {
    float dot[KCL];
    float x2 = 0.f;
#pragma unroll
    for (int k = 0; k < KCL; ++k) dot[k] = 0.f;
    const float* xr = &xs[tid * XS_STRIDE];
    for (int d = 0; d < DDIM; d += 4) {
      float x0 = xr[d], x1 = xr[d + 1], x2v = xr[d + 2], x3 = xr[d + 3];
      x2 += x0 * x0 + x1 * x1 + x2v * x2v + x3 * x3;
#pragma unroll
      for (int k = 0; k < KCL; ++k) {
        const float* cr = &cs[k * DDIM + d];
        dot[k] += x0 * cr[0] + x1 * cr[1] + x2v * cr[2] + x3 * cr[3];
      }
    }
    float temp = temperature[0];
    float logit[KCL], m = -3.4e38f;
#pragma unroll
    for (int k = 0; k < KCL; ++k) {
      float dist = x2 - 2.f * dot[k] + c2s[k];
      logit[k] = -temp * scale[k] * dist;
      m = fmaxf(m, logit[k]);
    }
    float sum = 0.f, w[KCL];
#pragma unroll
    for (int k = 0; k < KCL; ++k) { w[k] = __expf(logit[k] - m); sum += w[k]; }
    float inv = 1.f / sum;
    float* wr = &ws[tid * WS_STRIDE];
#pragma unroll
    for (int k = 0; k < KCL; ++k) wr[k] = w[k] * inv;
#pragma unroll
    for (int k = KCL; k < WS_STRIDE; ++k) wr[k] = 0.f;   // pad M rows 8..15 with zeros
  }
  __syncthreads();

  // s_w partial: fixed-order serial sum (deterministic, no float atomics)
  if (tid < KCL) {
    float s = 0.f;
    for (int t = 0; t < TCH; ++t) s += ws[t * WS_STRIDE + tid];
    sw_part[(long)blockIdx.x * KCL + tid] = s;
  }
  // (ws/xs were written before the barrier above; WMMA below reads only those)

  // ---- WMMA accumulate: s_wx[k,d] += w^T * x , s_wx2[k,d] += w^T * x^2 ----
  // Each wave owns 64 d-columns (4 tiles of 16). V_WMMA_F32_16X16X4_F32:
  //   A (16x4 f32): lane<16 -> M=lane, VGPR r is K=r ; lane>=16 -> M=lane-16, K=r+2
  //   B (4x16 f32): assumed transposed-symmetric: VGPR r, lane l -> K=r+2*(l>>4), N=l&15
  //   C/D (16x16 f32): VGPR v, lane l -> M=v+8*(l>>4), N=l&15
  const int wave = tid >> 5, lane = tid & 31;
  const int lane16 = lane & 15, hi = lane >> 4;
  v8f accx[4]  = {};
  v8f accx2[4] = {};

  for (int tt = 0; tt < TCH; tt += 4) {
    v2f a;
    a[0] = ws[(tt + 2 * hi + 0) * WS_STRIDE + lane16];
    a[1] = ws[(tt + 2 * hi + 1) * WS_STRIDE + lane16];
#pragma unroll
    for (int j = 0; j < 4; ++j) {
      int d0 = wave * 64 + j * 16;
      float b0 = xs[(tt + 2 * hi + 0) * XS_STRIDE + d0 + lane16];
      float b1 = xs[(tt + 2 * hi + 1) * XS_STRIDE + d0 + lane16];
      v2f bv, bq;
      bv[0] = b0;      bv[1] = b1;
      bq[0] = b0 * b0; bq[1] = b1 * b1;
      accx[j]  = __builtin_amdgcn_wmma_f32_16x16x4_f32(
          false, a, false, bv, (short)0, accx[j],  false, false);
      accx2[j] = __builtin_amdgcn_wmma_f32_16x16x4_f32(
          false, a, false, bq, (short)0, accx2[j], false, false);
    }
  }

  // store partials: only M=0..7 (k rows) are meaningful -> lanes 0..15, VGPRs 0..7
  if (hi == 0) {
#pragma unroll
    for (int j = 0; j < 4; ++j) {
      int d0 = wave * 64 + j * 16;
#pragma unroll
      for (int v = 0; v < 8; ++v) {
        long o = ((long)blockIdx.x * KCL + v) * DDIM + d0 + lane16;
        swx_part[o]  = accx[j][v];
        swx2_part[o] = accx2[j][v];
      }
    }
  }
}

// ---------------- Pass 2: chunk reduction + mean/var + LayerNorm(512) ----------------
__global__ __launch_bounds__(256) void lde_pass2(
    const float* __restrict__ centers, const float* __restrict__ swx_part,
    const float* __restrict__ swx2_part, const float* __restrict__ sw_part,
    float* __restrict__ out)
{
  __shared__ float r1[256];
  __shared__ float r2[256];
  const int b = blockIdx.x >> 3, k = blockIdx.x & 7;
  const int d = threadIdx.x;

  float swx = 0.f, swx2 = 0.f, sw = 0.f;
  for (int ch = 0; ch < NCH; ++ch) {
    long cb = (long)b * NCH + ch;
    long base = (cb * KCL + k) * DDIM + d;
    swx  += swx_part[base];
    swx2 += swx2_part[base];
    sw   += sw_part[cb * KCL + k];
  }
  float c = centers[k * DDIM + d];
  float mean = swx - c * sw;
  float E    = swx2 - 2.f * c * swx + c * c * sw;
  float var  = E - mean * mean;

  // LayerNorm over the 512 stats {mean_0..255, var_0..255}
  r1[d] = mean + var;
  r2[d] = mean * mean + var * var;
  __syncthreads();
  for (int s = 128; s > 0; s >>= 1) {
    if (d < s) { r1[d] += r1[d + s]; r2[d] += r2[d + s]; }
    __syncthreads();
  }
  float mu  = r1[0] * (1.f / 512.f);
  float vv  = r2[0] * (1.f / 512.f) - mu * mu;
  float inv = rsqrtf(vv + 1e-5f);

  long ob = (long)b * (KCL * 2 * DDIM) + (long)k * (2 * DDIM);
  out[ob + d]        = (mean - mu) * inv;
  out[ob + DDIM + d] = (var  - mu) * inv;
}

// ---------------- launch ----------------
extern "C" void kernel_launch(void* const* d_in, const int* in_sizes, int n_in,
                              void* d_out, int out_size, void* d_ws, size_t ws_size,
                              hipStream_t stream) {
  (void)in_sizes; (void)n_in; (void)out_size; (void)ws_size;
  const float* x           = (const float*)d_in[0];
  const float* centers     = (const float*)d_in[1];
  const float* scale       = (const float*)d_in[2];
  const float* temperature = (const float*)d_in[3];
  float* out = (float*)d_out;

  // workspace: swx_part (16*16*8*256) | swx2_part (same) | sw_part (16*16*8)
  float* swx_part  = (float*)d_ws;
  float* swx2_part = swx_part + (size_t)16 * NCH * KCL * DDIM;
  float* sw_part   = swx2_part + (size_t)16 * NCH * KCL * DDIM;

  size_t smem = (size_t)(TCH * XS_STRIDE + TCH * WS_STRIDE + KCL * DDIM + KCL) * sizeof(float);
  lde_pass1<<<16 * NCH, 128, smem, stream>>>(x, centers, scale, temperature,
                                             swx_part, swx2_part, sw_part);
  lde_pass2<<<16 * KCL, 256, 0, stream>>>(centers, swx_part, swx2_part, sw_part, out);
}